// QNN_1975684956466
// MI455X (gfx1250) — compile-verified
//
#include <hip/hip_runtime.h>
#include <hip/hip_bf16.h>
#include <math.h>

// ---------------------------------------------------------------------------
// QNN circuit, analytically collapsed:
//   z_q        = cos(w[q,1]) * cos(x[b,q] + w[q,0])   (per-qubit <Z>, product state)
//   Z_out[0]   = prod_{q=1..19} z_q                   (CNOT-ring XOR structure)
//   Z_out[i>0] = prod_{q=0..i}  z_q
//   logits     = Z_out @ fc_w^T + fc_b   -> FP32 WMMA 16x16x4, 5 K-chunks
//   out        = log_softmax(logits)
// Single wave32, single workgroup; EXEC all-ones at every WMMA.
// Operands are staged zero-padded in LDS so fragment gathers are unconditional
// ds_load_b64 (no per-load exec masking).
// ---------------------------------------------------------------------------

typedef __attribute__((ext_vector_type(2))) float v2f;
typedef __attribute__((ext_vector_type(8))) float v8f;

#define NQ    20
#define BATCH 4

__global__ __launch_bounds__(32, 1)
void qnn_analytic_wmma(const float* __restrict__ x,     // (BATCH, NQ)
                       const float* __restrict__ w,     // (NQ, 3)
                       const float* __restrict__ fcw,   // (3, NQ)
                       const float* __restrict__ fcb,   // (3,)
                       float* __restrict__ out)         // (BATCH, 3)
{
    __shared__ float zb[BATCH][NQ];   // <Z_q> before CNOT ring
    __shared__ float Apad[16][NQ];    // zero-padded A: rows 0..3 = Z_out rows
    __shared__ float Bpad[16][NQ];    // zero-padded B: rows 0..2 = fc_w rows
    __shared__ float lg[BATCH][4];    // logits (padded)

    const int lane = threadIdx.x;

    // ---- Step 0: zero-fill padded operand tiles (16*20 floats each) ----
    #pragma unroll
    for (int r = 0; r < 10; ++r) {
        int idx = lane + 32 * r;                 // 0..319
        (&Apad[0][0])[idx] = 0.0f;
        (&Bpad[0][0])[idx] = 0.0f;
    }

    // ---- Step 1: per-qubit expectation values + stage fc_w rows ----
    #pragma unroll
    for (int r = 0; r < 3; ++r) {
        int idx = lane + 32 * r;
        if (idx < BATCH * NQ) {
            int b = idx / NQ;
            int q = idx - b * NQ;
            float th = x[b * NQ + q] + w[q * 3 + 0];   // RX(x)*RX(w0) = RX(x+w0)
            zb[b][q] = cosf(w[q * 3 + 1]) * cosf(th);  // RZ drops out of <Z>
        }
        if (idx < 3 * NQ) {
            (&Bpad[0][0])[idx] = fcw[idx];             // Bpad[n][q] = fc_w[n][q]
        }
    }
    __syncthreads();

    // ---- Step 2: CNOT-ring -> prefix products (one batch row per lane) ----
    if (lane < BATCH) {
        float p = 1.0f;
        #pragma unroll
        for (int q = 1; q < NQ; ++q) p *= zb[lane][q];
        Apad[lane][0] = p;                     // Z_out[0] = prod z[1..19]
        p = zb[lane][0];
        #pragma unroll
        for (int q = 1; q < NQ; ++q) {
            p *= zb[lane][q];
            Apad[lane][q] = p;                 // Z_out[i] = prod z[0..i]
        }
    }
    __syncthreads();

    // ---- Step 3: FC head via FP32 WMMA: D(16x16) += A(16x4) x B(4x16) ----
    // ISA 16x4 f32 A layout: lanes 0-15 hold K=kb+{0,1}, lanes 16-31 K=kb+{2,3};
    // B (4x16) mirrored. Rows >= BATCH / cols >= 3 are zero via padding, so the
    // fragment loads are unconditional 8-byte LDS loads.
    v8f acc = {0.f, 0.f, 0.f, 0.f, 0.f, 0.f, 0.f, 0.f};
    const int m    = lane & 15;
    const int koff = (lane >> 4) << 1;          // 0 or 2
    #pragma unroll
    for (int kb = 0; kb < NQ; kb += 4) {
        v2f a  = *(const v2f*)&Apad[m][kb + koff];
        v2f bf = *(const v2f*)&Bpad[m][kb + koff];
        // (neg_a, A, neg_b, B, c_mod, C, reuse_a, reuse_b)
        acc = __builtin_amdgcn_wmma_f32_16x16x4_f32(
                  false, a, false, bf, (short)0, acc, false, false);
    }
    // D layout: lanes 0-15 hold N=lane, VGPR r -> row M=r. We need M=0..3, N=0..2.
    if (lane < 3) {
        float bias = fcb[lane];
        #pragma unroll
        for (int b = 0; b < BATCH; ++b) lg[b][lane] = acc[b] + bias;
    }
    __syncthreads();

    // ---- Step 4: log_softmax over the 3 classes, one row per lane ----
    if (lane < BATCH) {
        float l0 = lg[lane][0], l1 = lg[lane][1], l2 = lg[lane][2];
        float mx  = fmaxf(l0, fmaxf(l1, l2));
        float s   = expf(l0 - mx) + expf(l1 - mx) + expf(l2 - mx);
        float lse = mx + logf(s);
        out[lane * 3 + 0] = l0 - lse;
        out[lane * 3 + 1] = l1 - lse;
        out[lane * 3 + 2] = l2 - lse;
    }
}

extern "C" void kernel_launch(void* const* d_in, const int* in_sizes, int n_in,
                              void* d_out, int out_size, void* d_ws, size_t ws_size,
                              hipStream_t stream) {
    const float* x   = (const float*)d_in[0];  // (4, 20)
    const float* w   = (const float*)d_in[1];  // (20, 3)
    const float* fcw = (const float*)d_in[2];  // (3, 20)
    const float* fcb = (const float*)d_in[3];  // (3,)
    float* out = (float*)d_out;                // (4, 3)
    (void)in_sizes; (void)n_in; (void)out_size; (void)d_ws; (void)ws_size;
    qnn_analytic_wmma<<<1, 32, 0, stream>>>(x, w, fcw, fcb, out);
}